// FrameEnergyLoss_12146167513820
// MI455X (gfx1250) — compile-verified
//
#include <hip/hip_runtime.h>
#include <hip/hip_bf16.h>

typedef float v2f __attribute__((ext_vector_type(2)));
typedef float v8f __attribute__((ext_vector_type(8)));

#define BLK       256
#define NBLK_NODE 1024
#define NBLK_ELEM 1024

__device__ __forceinline__ float blockReduceSum(float v) {
    __shared__ float sm[BLK];
    const int t = threadIdx.x;
    sm[t] = v;
    __syncthreads();
    for (int off = BLK / 2; off > 0; off >>= 1) {
        if (t < off) sm[t] += sm[t + off];
        __syncthreads();
    }
    return sm[0];
}

// Kernel A: per-node u_phys = pred_raw * [u_c, u_c, theta_c][batch]; partial W_ext = sum(F_ext . u_phys)
__global__ void fel_node_kernel(const float* __restrict__ pred_raw,
                                const float* __restrict__ u_c,
                                const float* __restrict__ theta_c,
                                const int*   __restrict__ batch,
                                const float* __restrict__ F_ext,
                                float* __restrict__ u_phys,
                                float* __restrict__ partW,
                                int n_nodes) {
    float acc = 0.0f;
    const int stride = gridDim.x * blockDim.x;
    for (int i = blockIdx.x * blockDim.x + threadIdx.x; i < n_nodes; i += stride) {
        const int   b  = batch[i];
        const float uc = u_c[b];
        const float tc = theta_c[b];
        const float p0 = pred_raw[3 * i + 0];
        const float p1 = pred_raw[3 * i + 1];
        const float p2 = pred_raw[3 * i + 2];
        const float up0 = p0 * uc;
        const float up1 = p1 * uc;
        const float up2 = p2 * tc;
        u_phys[3 * i + 0] = up0;
        u_phys[3 * i + 1] = up1;
        u_phys[3 * i + 2] = up2;
        acc += F_ext[3 * i + 0] * up0 + F_ext[3 * i + 1] * up1 + F_ext[3 * i + 2] * up2;
    }
    const float s = blockReduceSum(acc);
    if (threadIdx.x == 0) partW[blockIdx.x] = s;
}

// Kernel B: per-element strain energy via the closed-form quadratic of the 6x6 beam stiffness.
__global__ void fel_elem_kernel(const int2*  __restrict__ conn,
                                const float* __restrict__ dir,
                                const float* __restrict__ elen,
                                const float* __restrict__ pE,
                                const float* __restrict__ pA,
                                const float* __restrict__ pI,
                                const float* __restrict__ u_phys,
                                float* __restrict__ partU,
                                int n_elem) {
    float acc = 0.0f;
    const int stride = gridDim.x * blockDim.x;
    for (int e = blockIdx.x * blockDim.x + threadIdx.x; e < n_elem; e += stride) {
        // Speculative prefetch of next iteration's streamed operands (global_prefetch_b8).
        __builtin_prefetch(&conn[e + stride], 0, 3);
        __builtin_prefetch(&dir[3 * (e + stride)], 0, 3);
        __builtin_prefetch(&elen[e + stride], 0, 3);

        const int2  cn = conn[e];
        const float c  = dir[3 * e + 0];
        const float s  = dir[3 * e + 2];

        const float a0 = u_phys[3 * cn.x + 0];
        const float a1 = u_phys[3 * cn.x + 1];
        const float a2 = u_phys[3 * cn.x + 2];
        const float b0 = u_phys[3 * cn.y + 0];
        const float b1 = u_phys[3 * cn.y + 1];
        const float b2 = u_phys[3 * cn.y + 2];

        const float uA =  c * a0 + s * a1;
        const float wA = -s * a0 + c * a1;
        const float tA = -a2;
        const float uB =  c * b0 + s * b1;
        const float wB = -s * b0 + c * b1;
        const float tB = -b2;

        const float L   = elen[e];
        const float E   = pE[e];
        const float ea  = E * pA[e];
        const float ei  = E * pI[e];
        const float eaL = ea / L;
        const float ei1 = ei / L;
        const float ei2 = ei1 / L;
        const float ei3 = ei2 / L;

        const float du = uA - uB;
        const float dw = wA - wB;
        const float ts = tA + tB;

        const float U = eaL * du * du
                      + 12.0f * ei3 * dw * dw
                      + 12.0f * ei2 * dw * ts
                      + 4.0f  * ei1 * (tA * tA + tA * tB + tB * tB);
        acc += 0.5f * U;
    }
    const float su = blockReduceSum(acc);
    if (threadIdx.x == 0) partU[blockIdx.x] = su;
}

// Kernel C: E_c_total = sum(max(F_c*u_c, 1e-30)) over graphs (single block).
__global__ void fel_ec_kernel(const float* __restrict__ F_c,
                              const float* __restrict__ u_c,
                              float* __restrict__ ec,
                              int n_graphs) {
    float acc = 0.0f;
    for (int i = threadIdx.x; i < n_graphs; i += blockDim.x)
        acc += fmaxf(F_c[i] * u_c[i], 1e-30f);
    const float s = blockReduceSum(acc);
    if (threadIdx.x == 0) ec[0] = s;
}

// Kernel D: single-wave epilogue. Accumulate the 2x1024 partials into a 16x4 FP32
// A-operand (64 lanes' worth of accumulators), then V_WMMA_F32_16X16X4_F32 with an
// all-ones 4x16 B reduces 64 -> 16 row sums in one matrix op; lanes 0/16 of the
// D tile hold M=0..7 / M=8..15, one shuffle finishes the scalar.
__global__ void fel_final_kernel(const float* __restrict__ partU,
                                 const float* __restrict__ partW,
                                 const float* __restrict__ ec,
                                 float* __restrict__ out,
                                 int n_part, float inv_ngraphs) {
    const int lane = threadIdx.x;  // 32 threads, one wave, EXEC all ones
    float acc0 = 0.0f, acc1 = 0.0f;
    for (int i = lane; i < n_part; i += 64)      acc0 += partU[i] - partW[i];
    for (int i = lane + 32; i < n_part; i += 64) acc1 += partU[i] - partW[i];

    v2f A; A.x = acc0; A.y = acc1;
    v2f B; B.x = 1.0f; B.y = 1.0f;
    v8f C = {0.0f, 0.0f, 0.0f, 0.0f, 0.0f, 0.0f, 0.0f, 0.0f};
    // 8-arg pattern: (neg_a, A, neg_b, B, c_mod, C, reuse_a, reuse_b)
    v8f D = __builtin_amdgcn_wmma_f32_16x16x4_f32(false, A, false, B, (short)0, C,
                                                  false, false);
    float s = D[0] + D[1] + D[2] + D[3] + D[4] + D[5] + D[6] + D[7];
    const float other = __shfl(s, 16, 32);
    if (lane == 0) {
        const float Pi = s + other;           // lane0: rows 0..7, lane16: rows 8..15
        out[0] = Pi / ec[0] * inv_ngraphs;
    }
}

extern "C" void kernel_launch(void* const* d_in, const int* in_sizes, int n_in,
                              void* d_out, int out_size, void* d_ws, size_t ws_size,
                              hipStream_t stream) {
    const float* pred_raw = (const float*)d_in[0];
    const float* u_c      = (const float*)d_in[1];
    const float* theta_c  = (const float*)d_in[2];
    const float* F_c      = (const float*)d_in[3];
    const int2*  conn     = (const int2*) d_in[4];
    const float* elen     = (const float*)d_in[5];
    const float* pE       = (const float*)d_in[6];
    const float* pA       = (const float*)d_in[7];
    const float* pI       = (const float*)d_in[8];
    const float* dir      = (const float*)d_in[9];
    const float* F_ext    = (const float*)d_in[10];
    const int*   batch    = (const int*)  d_in[11];

    const int n_nodes  = in_sizes[11];
    const int n_elem   = in_sizes[5];
    const int n_graphs = in_sizes[1];

    float* ws     = (float*)d_ws;
    float* u_phys = ws;                         // 3*n_nodes floats
    float* partW  = u_phys + (size_t)3 * n_nodes;  // NBLK_NODE
    float* partU  = partW + NBLK_NODE;             // NBLK_ELEM
    float* ec     = partU + NBLK_ELEM;             // 1

    fel_node_kernel<<<NBLK_NODE, BLK, 0, stream>>>(pred_raw, u_c, theta_c, batch,
                                                   F_ext, u_phys, partW, n_nodes);
    fel_elem_kernel<<<NBLK_ELEM, BLK, 0, stream>>>(conn, dir, elen, pE, pA, pI,
                                                   u_phys, partU, n_elem);
    fel_ec_kernel<<<1, BLK, 0, stream>>>(F_c, u_c, ec, n_graphs);
    fel_final_kernel<<<1, 32, 0, stream>>>(partU, partW, ec, (float*)d_out,
                                           NBLK_ELEM, 1.0f / (float)n_graphs);
}